// CustomLoss_39779987096184
// MI455X (gfx1250) — compile-verified
//
#include <hip/hip_runtime.h>
#include <hip/hip_bf16.h>

#define N_SAMPLES 8192
#define D_DIM     3072
#define NCHUNKS   32           // N_SAMPLES / ROWS_PER_CHUNK
#define ROWS_PER_CHUNK 256
#define LOG_2PI   1.8378770664093453f

typedef __attribute__((ext_vector_type(2))) float v2f;
typedef __attribute__((ext_vector_type(8))) float v8f;

#if defined(__has_builtin)
#  if __has_builtin(__builtin_amdgcn_wmma_f32_16x16x4_f32)
#    define HAVE_WMMA_F32X4 1
#  endif
#endif
#ifndef HAVE_WMMA_F32X4
#  define HAVE_WMMA_F32X4 0
#endif

// ---------------------------------------------------------------- kernel 0
// Class counts (deterministic integer reduction, one block).
__global__ __launch_bounds__(256) void k_counts(const int* __restrict__ tgt,
                                                float* __restrict__ cnt) {
    __shared__ int s0[256], s1[256];
    int t = threadIdx.x;
    int c0 = 0, c1 = 0;
    for (int i = t; i < N_SAMPLES; i += 256) {
        int v = tgt[i];
        c0 += (v == 0);
        c1 += (v == 1);
    }
    s0[t] = c0; s1[t] = c1;
    __syncthreads();
    for (int off = 128; off > 0; off >>= 1) {
        if (t < off) { s0[t] += s0[t + off]; s1[t] += s1[t + off]; }
        __syncthreads();
    }
    if (t == 0) { cnt[0] = (float)s0[0]; cnt[1] = (float)s1[0]; }
}

// ---------------------------------------------------------------- kernel 1
// Per-class column sums of mean & log_sd via V_WMMA_F32_16X16X4_F32:
//   S = onehot^T (2 x N)  @  X (N x D)
// A (16x4): rows = class one-hot built from target; B (4x16): 4 sample rows
// x 16 columns of X. Each wave owns one 16-column slab and one 256-row chunk.
// partial layout: [chunk][comp][D], comp: 0=mean_c0 1=mean_c1 2=lsd_c0 3=lsd_c1
__global__ __launch_bounds__(256) void k_class_colsum(
        const float* __restrict__ mean, const float* __restrict__ lsd,
        const int* __restrict__ tgt, float* __restrict__ partial) {
    const int lane = threadIdx.x & 31;
    const int wave = threadIdx.x >> 5;
    const int d0   = (blockIdx.x * 8 + wave) * 16;
    const int nb   = blockIdx.y * ROWS_PER_CHUNK;
    const int m    = lane & 15;      // M index (class row for M<2)
    const int hi   = lane >> 4;      // lane half selects K pair

    v8f accm = {};
    v8f accl = {};
    const size_t col = (size_t)d0 + m;

#pragma unroll 4
    for (int k0 = 0; k0 < ROWS_PER_CHUNK; k0 += 4) {
        const int n0 = nb + k0;
        // uniform -> scalar loads
        const int t0 = tgt[n0 + 0], t1 = tgt[n0 + 1];
        const int t2 = tgt[n0 + 2], t3 = tgt[n0 + 3];

        // A layout (32-bit 16x4): lanes 0-15: vgpr0=K0, vgpr1=K1;
        //                         lanes16-31: vgpr0=K2, vgpr1=K3
        v2f a;
        a[0] = (m == (hi ? t2 : t0)) ? 1.0f : 0.0f;
        a[1] = (m == (hi ? t3 : t1)) ? 1.0f : 0.0f;

        // B (4x16): mirror K mapping; n = lane%16
        const size_t r0 = (size_t)(n0 + (hi ? 2 : 0)) * D_DIM + col;
        const size_t r1 = (size_t)(n0 + (hi ? 3 : 1)) * D_DIM + col;
        v2f bm, bl;
        bm[0] = mean[r0]; bm[1] = mean[r1];
        bl[0] = lsd[r0];  bl[1] = lsd[r1];

#if HAVE_WMMA_F32X4
        accm = __builtin_amdgcn_wmma_f32_16x16x4_f32(
                   false, a, false, bm, (short)0, accm, false, false);
        accl = __builtin_amdgcn_wmma_f32_16x16x4_f32(
                   false, a, false, bl, (short)0, accl, false, false);
#else
        // VALU fallback: each lane-half accumulates its two rows
        accm[0] += a[0] * ((hi ? t2 : t0) == m && m == 0 ? bm[0] : (m == 0 && (hi ? t2 : t0) == 0 ? bm[0] : 0.f));
        accm[0] += ((hi ? t2 : t0) == 0 ? bm[0] : 0.f) + ((hi ? t3 : t1) == 0 ? bm[1] : 0.f);
        accm[1] += ((hi ? t2 : t0) == 1 ? bm[0] : 0.f) + ((hi ? t3 : t1) == 1 ? bm[1] : 0.f);
        accl[0] += ((hi ? t2 : t0) == 0 ? bl[0] : 0.f) + ((hi ? t3 : t1) == 0 ? bl[1] : 0.f);
        accl[1] += ((hi ? t2 : t0) == 1 ? bl[0] : 0.f) + ((hi ? t3 : t1) == 1 ? bl[1] : 0.f);
#endif
    }

#if !HAVE_WMMA_F32X4
    // combine lane halves (K2,K3 partials live in lanes 16-31)
    accm[0] += __shfl_down(accm[0], 16, 32);
    accm[1] += __shfl_down(accm[1], 16, 32);
    accl[0] += __shfl_down(accl[0], 16, 32);
    accl[1] += __shfl_down(accl[1], 16, 32);
#endif

    // C/D layout: vgpr r, lanes 0-15 hold M=r, N=lane -> class sums for
    // this column slab sit in accX[0] (class 0) / accX[1] (class 1), lanes 0-15.
    if (lane < 16) {
        float* p = partial + (size_t)blockIdx.y * 4 * D_DIM + d0 + lane;
        p[0 * D_DIM] = accm[0];
        p[1 * D_DIM] = accm[1];
        p[2 * D_DIM] = accl[0];
        p[3 * D_DIM] = accl[1];
    }
}

// ---------------------------------------------------------------- kernel 1b
// Reduce chunk partials -> mus/lsds (to d_out AND aligned ws copy) + invvar.
__global__ __launch_bounds__(256) void k_reduce_params(
        const float* __restrict__ partial, const float* __restrict__ cnt,
        float* __restrict__ out, float* __restrict__ params,
        float* __restrict__ invvar) {
    int i = blockIdx.x * 256 + threadIdx.x;           // 0 .. 4*D-1
    if (i >= 4 * D_DIM) return;
    int comp = i / D_DIM;
    int d    = i - comp * D_DIM;
    float s = 0.f;
    for (int c = 0; c < NCHUNKS; ++c)
        s += partial[((size_t)c * 4 + comp) * D_DIM + d];
    float v = s / cnt[comp & 1];
    out[1 + (size_t)comp * D_DIM + d] = v;   // mus[0],mus[1],lsds[0],lsds[1]
    params[(size_t)comp * D_DIM + d]  = v;   // 16B-aligned copy for pass 2
    if (comp >= 2)
        invvar[(size_t)(comp - 2) * D_DIM + d] = expf(-2.0f * v);
}

// ---------------------------------------------------------------- kernel 2
// logp[n] = -0.5*log2pi*D - sum_d( lsd[c,d] + 0.5*(z-mu)^2 * invvar[c,d] )
// One wave per row; float4 streaming loads (512B per wave per iteration).
__global__ __launch_bounds__(256) void k_logp(
        const float* __restrict__ z, const int* __restrict__ tgt,
        const float* __restrict__ params, const float* __restrict__ invvar,
        float* __restrict__ logp_out) {
    const int lane = threadIdx.x & 31;
    const int wave = threadIdx.x >> 5;
    const int n    = blockIdx.x * 8 + wave;
    const int c    = tgt[n];                 // uniform per wave -> scalar

    const float4* zr = (const float4*)(z      + (size_t)n * D_DIM);
    const float4* mu = (const float4*)(params + (size_t)c * D_DIM);
    const float4* ls = (const float4*)(params + (size_t)(2 + c) * D_DIM);
    const float4* iv = (const float4*)(invvar + (size_t)c * D_DIM);

    float acc = 0.f;
#pragma unroll 4
    for (int j = lane; j < D_DIM / 4; j += 32) {
        float4 zv = zr[j], mv = mu[j], lv = ls[j], vv = iv[j];
        float e0 = zv.x - mv.x;
        float e1 = zv.y - mv.y;
        float e2 = zv.z - mv.z;
        float e3 = zv.w - mv.w;
        acc += lv.x + 0.5f * e0 * e0 * vv.x;
        acc += lv.y + 0.5f * e1 * e1 * vv.y;
        acc += lv.z + 0.5f * e2 * e2 * vv.z;
        acc += lv.w + 0.5f * e3 * e3 * vv.w;
    }
    for (int off = 16; off > 0; off >>= 1)
        acc += __shfl_down(acc, off, 32);
    if (lane == 0)
        logp_out[n] = -0.5f * LOG_2PI * (float)D_DIM - acc;
}

// ---------------------------------------------------------------- kernel 3
// Per-class totals + prior (fixed-order, deterministic, one block).
__global__ __launch_bounds__(256) void k_final(
        const float* __restrict__ logp, const float* __restrict__ logdet,
        const int* __restrict__ tgt, const float* __restrict__ cnt,
        float* __restrict__ out) {
    __shared__ float s[4][256];
    int t = threadIdx.x;
    float lp0 = 0.f, lp1 = 0.f, ld0 = 0.f, ld1 = 0.f;
    for (int n = t; n < N_SAMPLES; n += 256) {
        int c = tgt[n];
        float lp = logp[n], ld = logdet[n];
        if (c == 0) { lp0 += lp; ld0 += ld; }
        else        { lp1 += lp; ld1 += ld; }
    }
    s[0][t] = lp0; s[1][t] = lp1; s[2][t] = ld0; s[3][t] = ld1;
    __syncthreads();
    for (int off = 128; off > 0; off >>= 1) {
        if (t < off)
            for (int q = 0; q < 4; ++q) s[q][t] += s[q][t + off];
        __syncthreads();
    }
    if (t == 0) {
        float c0 = cnt[0], c1 = cnt[1];
        float LP0 = s[0][0] / c0, LP1 = s[1][0] / c1;
        float LD0 = s[2][0] / c0, LD1 = s[3][0] / c1;
        out[20481] = LP0;
        out[20482] = LP1;
        out[0]     = 0.5f * ((LP0 + LD0) + (LP1 + LD1));
    }
}

extern "C" void kernel_launch(void* const* d_in, const int* in_sizes, int n_in,
                              void* d_out, int out_size, void* d_ws, size_t ws_size,
                              hipStream_t stream) {
    const float* z      = (const float*)d_in[0];
    const float* mean   = (const float*)d_in[1];
    const float* lsd    = (const float*)d_in[2];
    const float* logdet = (const float*)d_in[3];
    const int*   tgt    = (const int*)d_in[4];
    float* out = (float*)d_out;
    float* ws  = (float*)d_ws;

    // ws layout (floats): [0..1] counts, [4 .. 4+4D) params (aligned),
    // [4+4D .. 4+6D) invvar, [4+6D ..) chunk partials (NCHUNKS*4*D)
    float* cnt     = ws;
    float* params  = ws + 4;
    float* invvar  = ws + 4 + 4 * D_DIM;
    float* partial = ws + 4 + 6 * D_DIM;

    k_counts<<<1, 256, 0, stream>>>(tgt, cnt);

    dim3 g1(D_DIM / 128, N_SAMPLES / ROWS_PER_CHUNK);   // (24, 32)
    k_class_colsum<<<g1, 256, 0, stream>>>(mean, lsd, tgt, partial);

    k_reduce_params<<<(4 * D_DIM + 255) / 256, 256, 0, stream>>>(
        partial, cnt, out, params, invvar);

    k_logp<<<N_SAMPLES / 8, 256, 0, stream>>>(z, tgt, params, invvar,
                                              out + 12289);

    k_final<<<1, 256, 0, stream>>>(out + 12289, logdet, tgt, cnt, out);
}